// _DirectionalSSM_22316650070174
// MI455X (gfx1250) — compile-verified
//
#include <hip/hip_runtime.h>
#include <hip/hip_bf16.h>
#include <math.h>

// Problem constants (DirectionalSSM / Mamba selective scan)
#define B_  2
#define L_  4096
#define E_  2048
#define N_  16
#define K_  4
#define R_  64
#define O_  96              // R + 2N
#define M_  (B_ * L_)       // 8192 flattened (b,l)
#define NCH 16              // scan chunks
#define LC  (L_ / NCH)      // 256 steps per chunk
#define KSPLIT 4            // K-split for x_dbl GEMM
#define KCHUNK (E_ / KSPLIT)  // 512

typedef __attribute__((ext_vector_type(2))) float v2f;
typedef __attribute__((ext_vector_type(8))) float v8f;

// ---- workspace layout (in floats) ----
#define WS_XC    ((size_t)0)                          // (M_, E_)      x_conv
#define WS_XDBL  (WS_XC    + (size_t)M_ * E_)         // (M_, O_)      x_dbl
#define WS_DELTA (WS_XDBL  + (size_t)M_ * O_)         // (M_, E_)      delta
#define WS_P     (WS_DELTA + (size_t)M_ * E_)         // (B,NCH,E,N)   chunk dA products
#define WS_HLOC  (WS_P     + (size_t)B_ * NCH * E_ * N_)
#define WS_HINIT (WS_HLOC  + (size_t)B_ * NCH * E_ * N_)
#define WS_PART  (WS_HINIT + (size_t)B_ * NCH * E_ * N_)  // (KSPLIT, M_, O_)

// ---------------------------------------------------------------------------
// K1: depthwise causal conv1d (K=4) + bias + SiLU  -> x_conv
// ---------------------------------------------------------------------------
__global__ void conv_silu_kernel(const float* __restrict__ x,
                                 const float* __restrict__ w,     // (E,K)
                                 const float* __restrict__ bias,  // (E)
                                 float* __restrict__ xc) {
  size_t idx = (size_t)blockIdx.x * blockDim.x + threadIdx.x;  // over M_*E_
  int e = (int)(idx % E_);
  int m = (int)(idx / E_);
  int l = m % L_;
  float acc = bias[e];
#pragma unroll
  for (int k = 0; k < K_; ++k) {
    int lk = l - (K_ - 1) + k;
    if (lk >= 0)
      acc += x[(size_t)(m - (K_ - 1) + k) * E_ + e] * w[e * K_ + k];
  }
  xc[idx] = acc / (1.0f + __expf(-acc));   // silu
}

// ---------------------------------------------------------------------------
// K2: x_dbl[m,o] = sum_k x_conv[m,k] * x_proj_w[o,k]    (M=8192, O=96, K=2048)
// One wave computes a full 16x96 output strip (6 WMMA accumulators): the A
// operand is loaded once per k-step and reused across all 6 B tiles, so
// x_conv is read exactly once. K is split 4 ways for occupancy; partials are
// reduced by a separate kernel in a fixed order (deterministic, no atomics).
//
// fp32 WMMA 16x16x4 operand layout (wave32):
//   A 16x4: lane(0..15)->M, vgpr0:K=klo, vgpr1:K=klo+1, klo = (lane>=16)?2:0
//   B 4x16: lane(0..15)->N (mirrored), same K split.
//   D 16x16: vgpr v -> M = v + 8*(lane>=16), N = lane&15.
// ---------------------------------------------------------------------------
__global__ void xproj_wmma_kernel(const float* __restrict__ xc,
                                  const float* __restrict__ wproj,  // (O_, E_)
                                  float* __restrict__ part) {       // (KSPLIT, M_, O_)
  const int wid  = threadIdx.x >> 5;
  const int lane = threadIdx.x & 31;
  const int w = blockIdx.x * (blockDim.x >> 5) + wid;
  const int ks = w & (KSPLIT - 1);
  const int mt = w >> 2;                 // 0..511
  const int m0 = mt * 16;
  const int row = lane & 15;
  const int klo = ((lane >> 4) << 1) + ks * KCHUNK;

  const float* aBase = xc    + (size_t)(m0 + row) * E_ + klo;
  const float* bBase = wproj + (size_t)row * E_ + klo;   // + j*16*E_ per N-tile

  v8f acc[6] = {};
#pragma unroll 4
  for (int kk = 0; kk < KCHUNK; kk += 4) {
    v2f a = *(const v2f*)(aBase + kk);
#pragma unroll
    for (int j = 0; j < 6; ++j) {
      v2f b = *(const v2f*)(bBase + kk + (size_t)j * 16 * E_);
      acc[j] = __builtin_amdgcn_wmma_f32_16x16x4_f32(
          /*neg_a=*/false, a, /*neg_b=*/false, b,
          /*c_mod=*/(short)0, acc[j], /*reuse_a=*/false, /*reuse_b=*/false);
    }
  }

  float* dst = part + (size_t)ks * M_ * O_;
  const int mrow = m0 + (lane >> 4) * 8;
#pragma unroll
  for (int v = 0; v < 8; ++v) {
    const size_t o = (size_t)(mrow + v) * O_ + row;
#pragma unroll
    for (int j = 0; j < 6; ++j)
      dst[o + j * 16] = acc[j][v];
  }
}

// Deterministic fixed-order reduction of the 4 K-split partials.
__global__ void xdbl_reduce_kernel(const float* __restrict__ part,
                                   float* __restrict__ xdbl) {
  const size_t i = (size_t)blockIdx.x * blockDim.x + threadIdx.x;  // M_*O_
  float s = part[i];
#pragma unroll
  for (int ks = 1; ks < KSPLIT; ++ks)
    s += part[i + (size_t)ks * M_ * O_];
  xdbl[i] = s;
}

// ---------------------------------------------------------------------------
// K3: delta[m,e] = softplus( sum_r x_dbl[m,r] * dt_proj_w[e,r] + dt_proj_b[e] )
//     (M=8192, E=2048, K=64). One wave computes a 16x128 strip (8 WMMA
//     accumulators) so the A strip is read once per 8 output tiles.
// ---------------------------------------------------------------------------
__global__ void dtproj_softplus_wmma_kernel(const float* __restrict__ xdbl,
                                            const float* __restrict__ dtw,  // (E_, R_)
                                            const float* __restrict__ dtb,  // (E_)
                                            float* __restrict__ delta) {
  const int wid  = threadIdx.x >> 5;
  const int lane = threadIdx.x & 31;
  const int w = blockIdx.x * (blockDim.x >> 5) + wid;
  const int eg = w & 15;                 // 16 column groups of 128
  const int mt = w >> 4;                 // 0..511
  const int m0 = mt * 16;
  const int e0 = eg * 128;
  const int row = lane & 15;
  const int klo = (lane >> 4) << 1;

  const float* aBase = xdbl + (size_t)(m0 + row) * O_ + klo;
  const float* bBase = dtw  + (size_t)(e0 + row) * R_ + klo;  // + j*16*R_ per tile

  v8f acc[8] = {};
#pragma unroll 2
  for (int kk = 0; kk < R_; kk += 4) {
    v2f a = *(const v2f*)(aBase + kk);
#pragma unroll
    for (int j = 0; j < 8; ++j) {
      v2f b = *(const v2f*)(bBase + kk + (size_t)j * 16 * R_);
      acc[j] = __builtin_amdgcn_wmma_f32_16x16x4_f32(
          false, a, false, b, (short)0, acc[j], false, false);
    }
  }

  const int mrow = m0 + (lane >> 4) * 8;
#pragma unroll
  for (int j = 0; j < 8; ++j) {
    const int col = e0 + j * 16 + row;
    const float bb = dtb[col];
#pragma unroll
    for (int v = 0; v < 8; ++v) {
      float z = acc[j][v] + bb;
      float sp = (z > 20.0f) ? z : log1pf(__expf(z));
      delta[(size_t)(mrow + v) * E_ + col] = sp;
    }
  }
}

// ---------------------------------------------------------------------------
// K4: chunked scan pass 1 — per chunk, starting h=0, compute:
//     P[n]  = prod_t exp(delta_t * A[n])   (chunk dA product)
//     hloc  = chunk-final local state
// grid: (E_/256, NCH, B), block 256.  B_t staged via double-buffered LDS.
// ---------------------------------------------------------------------------
__global__ void scan_pass1_kernel(const float* __restrict__ delta,
                                  const float* __restrict__ xc,
                                  const float* __restrict__ xdbl,
                                  const float* __restrict__ A_log,  // (E,N)
                                  float* __restrict__ Pc,
                                  float* __restrict__ hloc) {
  const int e = blockIdx.x * blockDim.x + threadIdx.x;
  const int c = blockIdx.y;
  const int b = blockIdx.z;
  __shared__ float sB[2][N_];

  float A[N_], h[N_], P[N_];
#pragma unroll
  for (int n = 0; n < N_; ++n) {
    A[n] = -__expf(A_log[(size_t)e * N_ + n]);
    h[n] = 0.0f;
    P[n] = 1.0f;
  }

  const int l0 = c * LC;
  if (threadIdx.x < N_)
    sB[0][threadIdx.x] = xdbl[(size_t)(b * L_ + l0) * O_ + R_ + threadIdx.x];
  __syncthreads();

  for (int t = 0; t < LC; ++t) {
    const int l = l0 + t;
    const int buf = t & 1;
    if ((t + 1 < LC) && threadIdx.x < N_)
      sB[buf ^ 1][threadIdx.x] =
          xdbl[(size_t)(b * L_ + l + 1) * O_ + R_ + threadIdx.x];

    const size_t mi = (size_t)(b * L_ + l) * E_ + e;
    const float d  = delta[mi];
    const float du = d * xc[mi];
#pragma unroll
    for (int n = 0; n < N_; ++n) {
      float dA = __expf(d * A[n]);
      P[n] *= dA;
      h[n] = dA * h[n] + du * sB[buf][n];
    }
    __syncthreads();
  }

  const size_t base = (((size_t)b * NCH + c) * E_ + e) * N_;
#pragma unroll
  for (int n = 0; n < N_; ++n) {
    Pc[base + n]   = P[n];
    hloc[base + n] = h[n];
  }
}

// ---------------------------------------------------------------------------
// K5: sequential carry across the 16 chunks (per (b,e,n) state element)
// ---------------------------------------------------------------------------
__global__ void chunk_carry_kernel(const float* __restrict__ Pc,
                                   const float* __restrict__ hloc,
                                   float* __restrict__ Hinit) {
  const int idx = blockIdx.x * blockDim.x + threadIdx.x;  // B_*E_*N_
  const int n = idx % N_;
  const int e = (idx / N_) % E_;
  const int b = idx / (N_ * E_);
  float H = 0.0f;
  for (int c = 0; c < NCH; ++c) {
    const size_t base = (((size_t)b * NCH + c) * E_ + e) * N_ + n;
    Hinit[base] = H;
    H = Pc[base] * H + hloc[base];
  }
}

// ---------------------------------------------------------------------------
// K6: scan pass 2 — re-scan each chunk with the correct initial state,
//     emit y[b,l,e] = sum_n h[n]*C_t[n] + u*D[e].
// ---------------------------------------------------------------------------
__global__ void scan_pass2_kernel(const float* __restrict__ delta,
                                  const float* __restrict__ xc,
                                  const float* __restrict__ xdbl,
                                  const float* __restrict__ A_log,
                                  const float* __restrict__ Dvec,
                                  const float* __restrict__ Hinit,
                                  float* __restrict__ out) {
  const int e = blockIdx.x * blockDim.x + threadIdx.x;
  const int c = blockIdx.y;
  const int b = blockIdx.z;
  __shared__ float sB[2][N_];
  __shared__ float sC[2][N_];

  float A[N_], h[N_];
  const size_t hb = (((size_t)b * NCH + c) * E_ + e) * N_;
#pragma unroll
  for (int n = 0; n < N_; ++n) {
    A[n] = -__expf(A_log[(size_t)e * N_ + n]);
    h[n] = Hinit[hb + n];
  }
  const float Dv = Dvec[e];

  const int l0 = c * LC;
  if (threadIdx.x < 2 * N_) {
    const float vv = xdbl[(size_t)(b * L_ + l0) * O_ + R_ + threadIdx.x];
    if (threadIdx.x < N_) sB[0][threadIdx.x] = vv;
    else                  sC[0][threadIdx.x - N_] = vv;
  }
  __syncthreads();

  for (int t = 0; t < LC; ++t) {
    const int l = l0 + t;
    const int buf = t & 1;
    if ((t + 1 < LC) && threadIdx.x < 2 * N_) {
      const float vv = xdbl[(size_t)(b * L_ + l + 1) * O_ + R_ + threadIdx.x];
      if (threadIdx.x < N_) sB[buf ^ 1][threadIdx.x] = vv;
      else                  sC[buf ^ 1][threadIdx.x - N_] = vv;
    }

    const size_t mi = (size_t)(b * L_ + l) * E_ + e;
    const float d  = delta[mi];
    const float u  = xc[mi];
    const float du = d * u;
    float y = 0.0f;
#pragma unroll
    for (int n = 0; n < N_; ++n) {
      float dA = __expf(d * A[n]);
      h[n] = dA * h[n] + du * sB[buf][n];
      y += h[n] * sC[buf][n];
    }
    out[mi] = y + u * Dv;
    __syncthreads();
  }
}

// ---------------------------------------------------------------------------
extern "C" void kernel_launch(void* const* d_in, const int* in_sizes, int n_in,
                              void* d_out, int out_size, void* d_ws, size_t ws_size,
                              hipStream_t stream) {
  (void)in_sizes; (void)n_in; (void)out_size; (void)ws_size;
  const float* x_inner   = (const float*)d_in[0];  // (B,L,E)
  const float* conv_w    = (const float*)d_in[1];  // (E,K)
  const float* conv_b    = (const float*)d_in[2];  // (E)
  const float* x_proj_w  = (const float*)d_in[3];  // (O,E)
  const float* dt_proj_w = (const float*)d_in[4];  // (E,R)
  const float* dt_proj_b = (const float*)d_in[5];  // (E)
  const float* A_log     = (const float*)d_in[6];  // (E,N)
  const float* Dvec      = (const float*)d_in[7];  // (E)
  float* out = (float*)d_out;

  float* ws    = (float*)d_ws;
  float* xc    = ws + WS_XC;
  float* xdbl  = ws + WS_XDBL;
  float* delta = ws + WS_DELTA;
  float* Pc    = ws + WS_P;
  float* hloc  = ws + WS_HLOC;
  float* Hinit = ws + WS_HINIT;
  float* part  = ws + WS_PART;

  // K1: conv + silu
  conv_silu_kernel<<<(M_ * E_) / 256, 256, 0, stream>>>(x_inner, conv_w, conv_b, xc);

  // K2: x_dbl projection (WMMA, 16x96 strip per wave, K split 4 ways)
  {
    const int waves = (M_ / 16) * KSPLIT;   // 2048
    xproj_wmma_kernel<<<waves / 4, 128, 0, stream>>>(xc, x_proj_w, part);
    xdbl_reduce_kernel<<<(M_ * O_) / 256, 256, 0, stream>>>(part, xdbl);
  }

  // K3: dt projection + softplus (WMMA, 16x128 strip per wave)
  {
    const int waves = (M_ / 16) * (E_ / 128);  // 8192
    dtproj_softplus_wmma_kernel<<<waves / 4, 128, 0, stream>>>(xdbl, dt_proj_w,
                                                               dt_proj_b, delta);
  }

  // K4: chunk-local scans
  {
    dim3 grid(E_ / 256, NCH, B_);
    scan_pass1_kernel<<<grid, 256, 0, stream>>>(delta, xc, xdbl, A_log, Pc, hloc);
  }

  // K5: sequential carry across chunks
  chunk_carry_kernel<<<(B_ * E_ * N_) / 256, 256, 0, stream>>>(Pc, hloc, Hinit);

  // K6: final scans with correct initial states + output
  {
    dim3 grid(E_ / 256, NCH, B_);
    scan_pass2_kernel<<<grid, 256, 0, stream>>>(delta, xc, xdbl, A_log, Dvec,
                                                Hinit, out);
  }
}